// GATEncoder_44504451121829
// MI455X (gfx1250) — compile-verified
//
#include <hip/hip_runtime.h>
#include <hip/hip_bf16.h>

// ---------------------------------------------------------------------------
// GAT encoder for MI455X (gfx1250, wave32).
// GEMM uses V_WMMA_F32_16X16X4_F32 (full fp32 precision; GEMM is ~3 GFLOP
// total, far from the bottleneck). Edge phase is L2-resident (19 MB feature
// table << 192 MB L2) atomic-bound scatter; one wave per edge, 3 feats/lane.
// ---------------------------------------------------------------------------

#define GAT_N      50000
#define GAT_E      800000
#define GAT_INDIM  128
#define GAT_HID    96
#define GAT_HEADS  4
#define GAT_HD     24
#define NEG_SLOPE  0.2f

typedef __attribute__((ext_vector_type(2))) float v2f;
typedef __attribute__((ext_vector_type(8))) float v8f;

// --- monotonic float<->uint encoding for atomicMax-based segment max -------
__device__ __forceinline__ unsigned f2mono(float f) {
    unsigned u = __float_as_uint(f);
    return (u & 0x80000000u) ? ~u : (u | 0x80000000u);
}
__device__ __forceinline__ float mono2f(unsigned u) {
    u = (u & 0x80000000u) ? (u ^ 0x80000000u) : ~u;
    return __uint_as_float(u);
}

// --- generic zero fill -----------------------------------------------------
__global__ void k_zero(float* __restrict__ p, int n) {
    int i = blockIdx.x * blockDim.x + threadIdx.x;
    if (i < n) p[i] = 0.0f;
}

// --- WMMA fp32 GEMM: F[n, 96] = H[n, Fin] @ W[Fin, 96] ---------------------
// One wave computes a 16x16 tile of F. K stepped by 4 per WMMA.
// A 16x4 f32 layout : lanes 0-15 hold K=0(v0),K=1(v1); lanes 16-31 K=2,K=3.
// B 4x16 layout     : mirrored (row K striped across lanes within a VGPR).
// C/D 16x16 layout  : VGPR r -> M=r (lanes 0-15) / M=r+8 (lanes 16-31).
__global__ void k_gemm_wmma(const float* __restrict__ H,
                            const float* __restrict__ W,
                            float* __restrict__ F,
                            int nrows, int Fin) {
    int gtid   = blockIdx.x * blockDim.x + threadIdx.x;
    int warp   = gtid >> 5;                 // wave32
    int ntiles = (nrows >> 4) * (GAT_HID / 16);
    if (warp >= ntiles) return;             // wave-uniform guard: EXEC all-1s inside

    int tile_m = warp / (GAT_HID / 16);
    int tile_n = warp % (GAT_HID / 16);

    int lane = threadIdx.x & 31;
    int half = lane >> 4;                   // 0: K{0,1} / M=r   1: K{2,3} / M=r+8
    int l    = lane & 15;

    int arow = tile_m * 16 + l;             // A row owned by this lane
    int koff = half * 2;                    // A/B K offset within the 4-wide step
    int bcol = tile_n * 16 + l;             // B/C column owned by this lane

    const float* __restrict__ hrow = H + (size_t)arow * Fin;

    v8f c = {0.f, 0.f, 0.f, 0.f, 0.f, 0.f, 0.f, 0.f};
    for (int k = 0; k < Fin; k += 4) {
        v2f a, b;
        a.x = hrow[k + koff];
        a.y = hrow[k + koff + 1];
        b.x = W[(size_t)(k + koff)     * GAT_HID + bcol];
        b.y = W[(size_t)(k + koff + 1) * GAT_HID + bcol];
        // 8 args: (neg_a, A, neg_b, B, c_mod, C, reuse_a, reuse_b)
        c = __builtin_amdgcn_wmma_f32_16x16x4_f32(false, a, false, b,
                                                  (short)0, c, false, false);
    }
#pragma unroll
    for (int r = 0; r < 8; ++r) {
        int orow = tile_m * 16 + r + half * 8;
        F[(size_t)orow * GAT_HID + bcol] = c[r];
    }
}

// --- per-(node, head) attention logits el/er -------------------------------
__global__ void k_elr(const float* __restrict__ feat,
                      const float* __restrict__ al,
                      const float* __restrict__ ar,
                      float* __restrict__ el, float* __restrict__ er, int nh) {
    int i = blockIdx.x * blockDim.x + threadIdx.x;
    if (i >= nh) return;
    int n = i >> 2, h = i & 3;
    const float* fp  = feat + (size_t)n * GAT_HID + h * GAT_HD;
    const float* alp = al + h * GAT_HD;
    const float* arp = ar + h * GAT_HD;
    float sl = 0.f, sr = 0.f;
#pragma unroll
    for (int d = 0; d < GAT_HD; ++d) {
        float f = fp[d];
        sl += f * alp[d];
        sr += f * arp[d];
    }
    el[i] = sl;
    er[i] = sr;
}

__device__ __forceinline__ float edge_logit(const float* el, const float* er,
                                            int s, int d, int h) {
    float v = el[s * GAT_HEADS + h] + er[d * GAT_HEADS + h];
    return v > 0.f ? v : NEG_SLOPE * v;    // leaky_relu
}

// --- pass A: segment max over dst ------------------------------------------
__global__ void k_edge_max(const int* __restrict__ src, const int* __restrict__ dst,
                           const float* __restrict__ el, const float* __restrict__ er,
                           unsigned* __restrict__ m, int eh) {
    int i = blockIdx.x * blockDim.x + threadIdx.x;
    if (i >= eh) return;
    int e = i >> 2, h = i & 3;
    int s = src[e], d = dst[e];
    atomicMax(m + d * GAT_HEADS + h, f2mono(edge_logit(el, er, s, d, h)));
}

// --- pass B: segment sum of exp(e - m[dst]) --------------------------------
__global__ void k_edge_expsum(const int* __restrict__ src, const int* __restrict__ dst,
                              const float* __restrict__ el, const float* __restrict__ er,
                              const unsigned* __restrict__ m, float* __restrict__ denom,
                              int eh) {
    int i = blockIdx.x * blockDim.x + threadIdx.x;
    if (i >= eh) return;
    int e = i >> 2, h = i & 3;
    int s = src[e], d = dst[e];
    float v  = edge_logit(el, er, s, d, h);
    float ex = __expf(v - mono2f(m[d * GAT_HEADS + h]));
    atomicAdd(denom + d * GAT_HEADS + h, ex);
}

// --- pass C: out[dst] += alpha * feat[src] ; one wave per edge -------------
__global__ void k_edge_scatter(const int* __restrict__ src, const int* __restrict__ dst,
                               const float* __restrict__ el, const float* __restrict__ er,
                               const unsigned* __restrict__ m, const float* __restrict__ denom,
                               const float* __restrict__ feat, float* __restrict__ out,
                               int nedges) {
    int gtid = blockIdx.x * blockDim.x + threadIdx.x;
    int e = gtid >> 5;                      // wave32: one edge per wave
    if (e >= nedges) return;
    int lane = threadIdx.x & 31;
    int s = src[e], d = dst[e];
#pragma unroll
    for (int i = 0; i < 3; ++i) {
        int f = lane + 32 * i;              // 3 * 32 = 96 features
        int h = f / GAT_HD;
        float v     = edge_logit(el, er, s, d, h);
        float alpha = __expf(v - mono2f(m[d * GAT_HEADS + h])) / denom[d * GAT_HEADS + h];
        atomicAdd(out + (size_t)d * GAT_HID + f,
                  feat[(size_t)s * GAT_HID + f] * alpha);
    }
}

// --- bias + ELU (in place) --------------------------------------------------
__global__ void k_finalize(float* __restrict__ x, const float* __restrict__ b, int n) {
    int i = blockIdx.x * blockDim.x + threadIdx.x;
    if (i >= n) return;
    float v = x[i] + b[i % GAT_HID];
    x[i] = v > 0.f ? v : (__expf(v) - 1.0f);
}

// --- mean pooling over nodes: out[96] = mean_n h[n, :] ---------------------
__global__ void k_pool(const float* __restrict__ h, float* __restrict__ out, int nodes) {
    int f  = threadIdx.x;                   // blockDim.x == 96
    int n0 = blockIdx.x * 512;
    int n1 = n0 + 512;
    if (n1 > nodes) n1 = nodes;
    float s = 0.f;
    for (int n = n0; n < n1; ++n) s += h[(size_t)n * GAT_HID + f];
    atomicAdd(out + f, s * (1.0f / (float)nodes));
}

extern "C" void kernel_launch(void* const* d_in, const int* in_sizes, int n_in,
                              void* d_out, int out_size, void* d_ws, size_t ws_size,
                              hipStream_t stream) {
    const float* feats = (const float*)d_in[0];
    const int*   src   = (const int*)d_in[1];
    const int*   dst   = (const int*)d_in[2];

    const int N = in_sizes[0] / GAT_INDIM;  // 50000
    const int E = in_sizes[1];              // 800000
    const int NF = N * GAT_HID;             // node-feature elements
    const int N4 = N * GAT_HEADS;
    const int E4 = E * GAT_HEADS;

    // workspace carve-up (all fp32/u32)
    float* ws   = (float*)d_ws;
    float* hA   = ws;                       // N*96
    float* hB   = hA + (size_t)NF;          // N*96
    float* feat = hB + (size_t)NF;          // N*96
    float* el   = feat + (size_t)NF;        // N*4
    float* er   = el + (size_t)N4;          // N*4
    unsigned* m = (unsigned*)(er + (size_t)N4); // N*4  (monotonic-encoded max)
    float* den  = (float*)(m + (size_t)N4); // N*4

    const int B256 = 256;
    const int gNF  = (NF + B256 - 1) / B256;
    const int gN4  = (N4 + B256 - 1) / B256;
    const int gE4  = (E4 + B256 - 1) / B256;
    const int gSc  = ((E * 32) + B256 - 1) / B256;

    const float* hin = feats;
    int Fin = GAT_INDIM;
    float* hnext = hA;

    for (int l = 0; l < 3; ++l) {
        const float* W  = (const float*)d_in[3 + 4 * l];
        const float* al = (const float*)d_in[4 + 4 * l];
        const float* ar = (const float*)d_in[5 + 4 * l];
        const float* bb = (const float*)d_in[6 + 4 * l];

        // zero per-layer accumulators (m's encoded -inf is 0)
        k_zero<<<gN4, B256, 0, stream>>>((float*)m, N4);
        k_zero<<<gN4, B256, 0, stream>>>(den, N4);
        k_zero<<<gNF, B256, 0, stream>>>(hnext, NF);

        // dense GEMM via fp32 WMMA
        int warps  = (N / 16) * (GAT_HID / 16);
        int gGemm  = (warps * 32 + B256 - 1) / B256;
        k_gemm_wmma<<<gGemm, B256, 0, stream>>>(hin, W, feat, N, Fin);

        // attention logits
        k_elr<<<gN4, B256, 0, stream>>>(feat, al, ar, el, er, N4);

        // edge softmax + aggregation
        k_edge_max<<<gE4, B256, 0, stream>>>(src, dst, el, er, m, E4);
        k_edge_expsum<<<gE4, B256, 0, stream>>>(src, dst, el, er, m, den, E4);
        k_edge_scatter<<<gSc, B256, 0, stream>>>(src, dst, el, er, m, den, feat, hnext, E);

        // bias + ELU
        k_finalize<<<gNF, B256, 0, stream>>>(hnext, bb, NF);

        hin   = hnext;
        hnext = (l == 0) ? hB : hA;
        Fin   = GAT_HID;
    }

    // mean pool -> d_out[96]
    float* outp = (float*)d_out;
    k_zero<<<1, 128, 0, stream>>>(outp, out_size);
    int gPool = (N + 511) / 512;
    k_pool<<<gPool, GAT_HID, 0, stream>>>(hin, outp, N);
}